// FeaturesProcessing_11622181503192
// MI455X (gfx1250) — compile-verified
//
#include <hip/hip_runtime.h>
#include <hip/hip_bf16.h>

// MI455X / gfx1250, wave32. Complex per-pixel GEMM chain via V_WMMA_F32_16X16X4_F32,
// LDS-staged with coalesced global streaming of the 1.2 GB filter tensors.

typedef __attribute__((ext_vector_type(2))) float v2f;
typedef __attribute__((ext_vector_type(8))) float v8f;

#define NB   32      // batch
#define C1   64      // in channels
#define HID  128     // hidden channels
#define C2   64      // out channels
#define HW   4096    // 64*64 pixels
#define EPSV 1e-5f

#define PX      16   // pixels per workgroup
#define PXP     17   // padded pixel dim (bank-conflict-free: odd stride)
#define KC      8    // k-values staged per chunk (2 WMMA k-steps)
#define ASTRIDE 552  // floats per k-slice of A tile: 32*17 + 8 (half-wave offset
                     //  2*552 mod 64 == 16 == 17*16 -> disjoint bank sets)
#define BSTRIDE 272  // floats per k-slice of B tile: 16*17 (offset 32 -> disjoint)

static __device__ __forceinline__ v8f wmma_f32(v2f a, v2f b, v8f c) {
  // D = A(16x4,f32) * B(4x16,f32) + C(16x16,f32)
  // (neg_a, A, neg_b, B, c_mod, C, reuse_a, reuse_b); NEG[1:0] must be 0 for f32.
  return __builtin_amdgcn_wmma_f32_16x16x4_f32(false, a, false, b, (short)0, c,
                                               false, false);
}

static __device__ __forceinline__ float silu(float x) {
  return x * (1.0f / (1.0f + __expf(-x)));
}

// Complex WMMA macro-step: Cre += Are*Bre - Aim*Bim ; Cim += Are*Bim + Aim*Bre
static __device__ __forceinline__ void cwmma(v2f are, v2f aim, v2f bre, v2f bim,
                                             v8f& cre, v8f& cim) {
  v2f nim;
  nim.x = -aim.x;
  nim.y = -aim.y;
  cre = wmma_f32(are, bre, cre);
  cre = wmma_f32(nim, bim, cre);
  cim = wmma_f32(are, bim, cim);
  cim = wmma_f32(aim, bre, cim);
}

// ---------------------------------------------------------------------------
// Bias reduction: out[o,p] = sum_i b[i,o,p]   (complex interleaved)
// ---------------------------------------------------------------------------
__global__ void bias_sum_kernel(const float* __restrict__ b,
                                float* __restrict__ out, int ni, int no) {
  int idx = blockIdx.x * blockDim.x + threadIdx.x;   // o*HW + p
  if (idx >= no * HW) return;
  size_t stride = (size_t)no * HW * 2;
  const float* src = b + (size_t)idx * 2;
  float sr = 0.f, si = 0.f;
  for (int i = 0; i < ni; ++i) {
    sr += src[0];
    si += src[1];
    src += stride;
  }
  out[(size_t)idx * 2 + 0] = sr;
  out[(size_t)idx * 2 + 1] = si;
}

// ---------------------------------------------------------------------------
// Shared compute step: one staged K-chunk -> 2 complex WMMA k-steps for both
// M-tiles of this wave's pixel. Reads are bank-conflict-free (see strides).
// ---------------------------------------------------------------------------
static __device__ __forceinline__ void chunk_compute(
    const float* a_re, const float* a_im, const float* b_re,
    const float* b_im, int half, int l15, int px, v8f& cre0, v8f& cim0,
    v8f& cre1, v8f& cim1) {
#pragma unroll
  for (int ks = 0; ks < KC; ks += 4) {
    const int k = ks + half * 2;                       // k within chunk
    const float* arp = a_re + (size_t)k * ASTRIDE + px;
    const float* aip = a_im + (size_t)k * ASTRIDE + px;
    const float* brp = b_re + (size_t)k * BSTRIDE + px;
    const float* bip = b_im + (size_t)k * BSTRIDE + px;
    v2f are0, aim0, are1, aim1, bre, bim;
    are0.x = arp[l15 * PXP];
    are0.y = arp[ASTRIDE + l15 * PXP];
    aim0.x = aip[l15 * PXP];
    aim0.y = aip[ASTRIDE + l15 * PXP];
    are1.x = arp[(l15 + 16) * PXP];
    are1.y = arp[ASTRIDE + (l15 + 16) * PXP];
    aim1.x = aip[(l15 + 16) * PXP];
    aim1.y = aip[ASTRIDE + (l15 + 16) * PXP];
    bre.x = brp[l15 * PXP];
    bre.y = brp[BSTRIDE + l15 * PXP];
    bim.x = bip[l15 * PXP];
    bim.y = bip[BSTRIDE + l15 * PXP];
    cwmma(are0, aim0, bre, bim, cre0, cim0);
    cwmma(are1, aim1, bre, bim, cre1, cim1);
  }
}

// ---------------------------------------------------------------------------
// conv1: per pixel  y1[b,o] = sum_i x[b,i]*f1[i,o] + bias1s[o]   (complex)
// Grid: (HW/PX)*(HID/16) blocks of 512 threads (16 waves; wave = pixel).
// blockIdx = pg*8 + nt  (N-tiles of one pixel group adjacent -> L2 reuse of x).
// ---------------------------------------------------------------------------
__global__ __launch_bounds__(512) void conv1_kernel(
    const float* __restrict__ xr, const float* __restrict__ xi,
    const float* __restrict__ f1, const float* __restrict__ b1s,
    float* __restrict__ y1) {
  __shared__ float a_re[KC * ASTRIDE], a_im[KC * ASTRIDE];
  __shared__ float b_re[KC * BSTRIDE], b_im[KC * BSTRIDE];

  const int nt = blockIdx.x & 7;          // 8 N-tiles (HID=128)
  const int pg = blockIdx.x >> 3;
  const int p0 = pg * PX;
  const int n0 = nt * 16;
  const int t = threadIdx.x;
  const int lane = t & 31;
  const int px = t >> 5;                  // wave id = pixel in group
  const int half = lane >> 4;
  const int l15 = lane & 15;

  v8f cre0 = {}, cim0 = {}, cre1 = {}, cim1 = {};

  for (int c = 0; c < C1 / KC; ++c) {
    const int kb = c * KC;
    // --- coalesced global loads into registers (A: 64B rows, B: 128B rows) ---
    float ar[8], ai[8];
#pragma unroll
    for (int j = 0; j < 8; ++j) {
      const int e = t + j * 512;          // 8*32*16 elements per plane
      const int ph = e & 15, m = (e >> 4) & 31, kk = e >> 9;
      const size_t gi = ((size_t)m * C1 + kb + kk) * HW + p0 + ph;
      ar[j] = xr[gi];
      ai[j] = xi[gi];
    }
    float2 bv[4];
#pragma unroll
    for (int j = 0; j < 4; ++j) {
      const int e = t + j * 512;          // 8*16*16 complex elements
      const int ph = e & 15, nn = (e >> 4) & 15, kk = e >> 8;
      const size_t gi = (((size_t)(kb + kk) * HID + n0 + nn) * HW + p0 + ph) * 2;
      bv[j] = *(const float2*)(f1 + gi);
      if (c + 1 < C1 / KC)                 // prefetch next chunk's filter line
        __builtin_prefetch(f1 + gi + (size_t)KC * HID * HW * 2, 0, 0);
    }
    __syncthreads();                       // prior chunk's LDS reads complete
#pragma unroll
    for (int j = 0; j < 8; ++j) {
      const int e = t + j * 512;
      const int ph = e & 15, m = (e >> 4) & 31, kk = e >> 9;
      a_re[kk * ASTRIDE + m * PXP + ph] = ar[j];
      a_im[kk * ASTRIDE + m * PXP + ph] = ai[j];
    }
#pragma unroll
    for (int j = 0; j < 4; ++j) {
      const int e = t + j * 512;
      const int ph = e & 15, nn = (e >> 4) & 15, kk = e >> 8;
      b_re[kk * BSTRIDE + nn * PXP + ph] = bv[j].x;
      b_im[kk * BSTRIDE + nn * PXP + ph] = bv[j].y;
    }
    __syncthreads();
    chunk_compute(a_re, a_im, b_re, b_im, half, l15, px, cre0, cim0, cre1, cim1);
  }

  const int p = p0 + px;
  const int n = n0 + l15;
  const float2 bias = *(const float2*)(b1s + ((size_t)n * HW + p) * 2);
#pragma unroll
  for (int r = 0; r < 8; ++r) {
    const int mr = half * 8 + r;          // D layout: lanes 16-31 hold M+8
    float2 v0, v1;
    v0.x = cre0[r] + bias.x;
    v0.y = cim0[r] + bias.y;
    v1.x = cre1[r] + bias.x;
    v1.y = cim1[r] + bias.y;
    *(float2*)(y1 + (((size_t)mr * HID + n) * HW + p) * 2) = v0;
    *(float2*)(y1 + (((size_t)(mr + 16) * HID + n) * HW + p) * 2) = v1;
  }
}

// ---------------------------------------------------------------------------
// Instance-norm stats: per (b,ch) reduce over 4096 pixels; complex variance,
// complex inverse sqrt; fold scale/shift into affine (mul, add).
// ---------------------------------------------------------------------------
__global__ __launch_bounds__(256) void stats_kernel(
    const float* __restrict__ y, const float* __restrict__ scale,
    const float* __restrict__ shift, float* __restrict__ nparams, int nch) {
  const int bo = blockIdx.x;              // b*nch + o
  const int o = bo % nch;
  const int t = threadIdx.x;
  const float* base = y + (size_t)bo * HW * 2;

  float sr = 0.f, si = 0.f, qr = 0.f, qi = 0.f;
  for (int p = t; p < HW; p += 256) {
    const float2 v = ((const float2*)base)[p];
    sr += v.x;
    si += v.y;
    qr += v.x * v.x - v.y * v.y;          // complex square
    qi += 2.f * v.x * v.y;
  }

  __shared__ float red[4][256];
  red[0][t] = sr; red[1][t] = si; red[2][t] = qr; red[3][t] = qi;
  __syncthreads();
  for (int s = 128; s > 0; s >>= 1) {
    if (t < s) {
#pragma unroll
      for (int j = 0; j < 4; ++j) red[j][t] += red[j][t + s];
    }
    __syncthreads();
  }

  if (t == 0) {
    const float inv_n = 1.0f / (float)HW;
    const float mr = red[0][0] * inv_n;
    const float mi = red[1][0] * inv_n;
    float vr = red[2][0] * inv_n - (mr * mr - mi * mi);
    float vi = red[3][0] * inv_n - 2.f * mr * mi;
    vr += EPSV;
    // s = csqrt(vr + i*vi); inv = conj(s)/|s|^2, |s|^2 = |v|
    const float rad = sqrtf(vr * vr + vi * vi);
    const float u = sqrtf(fmaxf(0.f, (rad + vr) * 0.5f));
    const float w = copysignf(sqrtf(fmaxf(0.f, (rad - vr) * 0.5f)), vi);
    const float rr = (rad > 0.f) ? (1.0f / rad) : 0.f;
    const float ir = u * rr;
    const float ii = -w * rr;
    const float2 sc = ((const float2*)scale)[o];
    const float2 sh = ((const float2*)shift)[o];
    const float mul_re = ir * sc.x - ii * sc.y;
    const float mul_im = ir * sc.y + ii * sc.x;
    float4 q;
    q.x = mul_re;
    q.y = mul_im;
    q.z = sh.x - (mr * mul_re - mi * mul_im);
    q.w = sh.y - (mr * mul_im + mi * mul_re);
    ((float4*)nparams)[bo] = q;
  }
}

// ---------------------------------------------------------------------------
// conv2: A = silu(norm1(y1)) applied during staging; per pixel [32x128]x[128x64].
// Grid: (HW/PX)*(C2/16) blocks of 512 threads.
// ---------------------------------------------------------------------------
__global__ __launch_bounds__(512) void conv2_kernel(
    const float* __restrict__ y1, const float* __restrict__ n1,
    const float* __restrict__ f2, const float* __restrict__ b2s,
    float* __restrict__ y2) {
  __shared__ float a_re[KC * ASTRIDE], a_im[KC * ASTRIDE];
  __shared__ float b_re[KC * BSTRIDE], b_im[KC * BSTRIDE];

  const int nt = blockIdx.x & 3;          // 4 N-tiles (C2=64)
  const int pg = blockIdx.x >> 2;
  const int p0 = pg * PX;
  const int n0 = nt * 16;
  const int t = threadIdx.x;
  const int lane = t & 31;
  const int px = t >> 5;
  const int half = lane >> 4;
  const int l15 = lane & 15;

  v8f cre0 = {}, cim0 = {}, cre1 = {}, cim1 = {};

  for (int c = 0; c < HID / KC; ++c) {
    const int kb = c * KC;
    float ar[8], ai[8];
#pragma unroll
    for (int j = 0; j < 8; ++j) {
      const int e = t + j * 512;
      const int ph = e & 15, m = (e >> 4) & 31, kk = e >> 9;
      const float2 v =
          *(const float2*)(y1 + (((size_t)m * HID + kb + kk) * HW + p0 + ph) * 2);
      const float4 q = ((const float4*)n1)[(size_t)m * HID + kb + kk];
      ar[j] = silu(v.x * q.x - v.y * q.y + q.z);   // norm + swish fused
      ai[j] = silu(v.x * q.y + v.y * q.x + q.w);
    }
    float2 bv[4];
#pragma unroll
    for (int j = 0; j < 4; ++j) {
      const int e = t + j * 512;
      const int ph = e & 15, nn = (e >> 4) & 15, kk = e >> 8;
      const size_t gi = (((size_t)(kb + kk) * C2 + n0 + nn) * HW + p0 + ph) * 2;
      bv[j] = *(const float2*)(f2 + gi);
      if (c + 1 < HID / KC)
        __builtin_prefetch(f2 + gi + (size_t)KC * C2 * HW * 2, 0, 0);
    }
    __syncthreads();
#pragma unroll
    for (int j = 0; j < 8; ++j) {
      const int e = t + j * 512;
      const int ph = e & 15, m = (e >> 4) & 31, kk = e >> 9;
      a_re[kk * ASTRIDE + m * PXP + ph] = ar[j];
      a_im[kk * ASTRIDE + m * PXP + ph] = ai[j];
    }
#pragma unroll
    for (int j = 0; j < 4; ++j) {
      const int e = t + j * 512;
      const int ph = e & 15, nn = (e >> 4) & 15, kk = e >> 8;
      b_re[kk * BSTRIDE + nn * PXP + ph] = bv[j].x;
      b_im[kk * BSTRIDE + nn * PXP + ph] = bv[j].y;
    }
    __syncthreads();
    chunk_compute(a_re, a_im, b_re, b_im, half, l15, px, cre0, cim0, cre1, cim1);
  }

  const int p = p0 + px;
  const int n = n0 + l15;
  const float2 bias = *(const float2*)(b2s + ((size_t)n * HW + p) * 2);
#pragma unroll
  for (int r = 0; r < 8; ++r) {
    const int mr = half * 8 + r;
    float2 v0, v1;
    v0.x = cre0[r] + bias.x;
    v0.y = cim0[r] + bias.y;
    v1.x = cre1[r] + bias.x;
    v1.y = cim1[r] + bias.y;
    *(float2*)(y2 + (((size_t)mr * C2 + n) * HW + p) * 2) = v0;
    *(float2*)(y2 + (((size_t)(mr + 16) * C2 + n) * HW + p) * 2) = v1;
  }
}

// ---------------------------------------------------------------------------
// final: hx = x*fd + biasds ; out = ri_swish(hx + norm2(y2))   per pixel.
// Grid: (HW/PX)*(C2/16) blocks of 512 threads.
// ---------------------------------------------------------------------------
__global__ __launch_bounds__(512) void final_kernel(
    const float* __restrict__ xr, const float* __restrict__ xi,
    const float* __restrict__ fd, const float* __restrict__ bds,
    const float* __restrict__ y2, const float* __restrict__ n2,
    float* __restrict__ out) {
  __shared__ float a_re[KC * ASTRIDE], a_im[KC * ASTRIDE];
  __shared__ float b_re[KC * BSTRIDE], b_im[KC * BSTRIDE];

  const int nt = blockIdx.x & 3;
  const int pg = blockIdx.x >> 2;
  const int p0 = pg * PX;
  const int n0 = nt * 16;
  const int t = threadIdx.x;
  const int lane = t & 31;
  const int px = t >> 5;
  const int half = lane >> 4;
  const int l15 = lane & 15;

  v8f cre0 = {}, cim0 = {}, cre1 = {}, cim1 = {};

  for (int c = 0; c < C1 / KC; ++c) {
    const int kb = c * KC;
    float ar[8], ai[8];
#pragma unroll
    for (int j = 0; j < 8; ++j) {
      const int e = t + j * 512;
      const int ph = e & 15, m = (e >> 4) & 31, kk = e >> 9;
      const size_t gi = ((size_t)m * C1 + kb + kk) * HW + p0 + ph;
      ar[j] = xr[gi];
      ai[j] = xi[gi];
    }
    float2 bv[4];
#pragma unroll
    for (int j = 0; j < 4; ++j) {
      const int e = t + j * 512;
      const int ph = e & 15, nn = (e >> 4) & 15, kk = e >> 8;
      const size_t gi = (((size_t)(kb + kk) * C2 + n0 + nn) * HW + p0 + ph) * 2;
      bv[j] = *(const float2*)(fd + gi);
      if (c + 1 < C1 / KC)
        __builtin_prefetch(fd + gi + (size_t)KC * C2 * HW * 2, 0, 0);
    }
    __syncthreads();
#pragma unroll
    for (int j = 0; j < 8; ++j) {
      const int e = t + j * 512;
      const int ph = e & 15, m = (e >> 4) & 31, kk = e >> 9;
      a_re[kk * ASTRIDE + m * PXP + ph] = ar[j];
      a_im[kk * ASTRIDE + m * PXP + ph] = ai[j];
    }
#pragma unroll
    for (int j = 0; j < 4; ++j) {
      const int e = t + j * 512;
      const int ph = e & 15, nn = (e >> 4) & 15, kk = e >> 8;
      b_re[kk * BSTRIDE + nn * PXP + ph] = bv[j].x;
      b_im[kk * BSTRIDE + nn * PXP + ph] = bv[j].y;
    }
    __syncthreads();
    chunk_compute(a_re, a_im, b_re, b_im, half, l15, px, cre0, cim0, cre1, cim1);
  }

  const int p = p0 + px;
  const int n = n0 + l15;
  const float2 bias = *(const float2*)(bds + ((size_t)n * HW + p) * 2);
#pragma unroll
  for (int r = 0; r < 8; ++r) {
#pragma unroll
    for (int mt = 0; mt < 2; ++mt) {
      const int mr = mt * 16 + half * 8 + r;
      const size_t eidx = ((size_t)mr * C2 + n) * HW + p;
      const float2 yv = *(const float2*)(y2 + eidx * 2);
      const float4 q = ((const float4*)n2)[(size_t)mr * C2 + n];
      const float yn_re = yv.x * q.x - yv.y * q.y + q.z;
      const float yn_im = yv.x * q.y + yv.y * q.x + q.w;
      const float hr = (mt == 0 ? cre0[r] : cre1[r]) + bias.x;
      const float hi = (mt == 0 ? cim0[r] : cim1[r]) + bias.y;
      float2 v;
      v.x = silu(hr + yn_re);
      v.y = silu(hi + yn_im);
      *(float2*)(out + eidx * 2) = v;
    }
  }
}

// ---------------------------------------------------------------------------
extern "C" void kernel_launch(void* const* d_in, const int* in_sizes, int n_in,
                              void* d_out, int out_size, void* d_ws,
                              size_t ws_size, hipStream_t stream) {
  const float* xr     = (const float*)d_in[0];
  const float* xi     = (const float*)d_in[1];
  const float* f1     = (const float*)d_in[2];   // complex64 interleaved
  const float* b1     = (const float*)d_in[3];
  const float* scale1 = (const float*)d_in[4];
  const float* shift1 = (const float*)d_in[5];
  const float* f2     = (const float*)d_in[6];
  const float* b2     = (const float*)d_in[7];
  const float* scale2 = (const float*)d_in[8];
  const float* shift2 = (const float*)d_in[9];
  const float* fd     = (const float*)d_in[10];
  const float* bd     = (const float*)d_in[11];
  float* out = (float*)d_out;

  // Workspace layout (floats)
  float* ws  = (float*)d_ws;
  float* y1  = ws;                                  // 32*128*4096*2
  float* y2  = y1  + (size_t)NB * HID * HW * 2;
  float* b1s = y2  + (size_t)NB * C2 * HW * 2;
  float* b2s = b1s + (size_t)HID * HW * 2;
  float* bds = b2s + (size_t)C2 * HW * 2;
  float* n1  = bds + (size_t)C2 * HW * 2;
  float* n2  = n1  + (size_t)NB * HID * 4;

  // Phase 0: bias reductions (sum over input-channel axis)
  bias_sum_kernel<<<(HID * HW + 255) / 256, 256, 0, stream>>>(b1, b1s, C1, HID);
  bias_sum_kernel<<<(C2 * HW + 255) / 256, 256, 0, stream>>>(b2, b2s, HID, C2);
  bias_sum_kernel<<<(C2 * HW + 255) / 256, 256, 0, stream>>>(bd, bds, C1, C2);

  // Phase 1: conv1 (WMMA, LDS-staged) -> y1
  conv1_kernel<<<(HW / PX) * (HID / 16), 512, 0, stream>>>(xr, xi, f1, b1s, y1);

  // Phase 2: instance-norm stats for y1 -> affine params n1
  stats_kernel<<<NB * HID, 256, 0, stream>>>(y1, scale1, shift1, n1, HID);

  // Phase 3: conv2 (WMMA, norm+silu fused into staging) -> y2
  conv2_kernel<<<(HW / PX) * (C2 / 16), 512, 0, stream>>>(y1, n1, f2, b2s, y2);

  // Phase 4: instance-norm stats for y2 -> affine params n2
  stats_kernel<<<NB * C2, 256, 0, stream>>>(y2, scale2, shift2, n2, C2);

  // Phase 5: down-bneck conv (WMMA) + norm2(y2) residual + silu -> out
  final_kernel<<<(HW / PX) * (C2 / 16), 512, 0, stream>>>(xr, xi, fd, bds, y2,
                                                          n2, out);
}